// garGATConv_52871047413957
// MI455X (gfx1250) — compile-verified
//
#include <hip/hip_runtime.h>
#include <math.h>

// Problem constants (from reference)
#define N_NODES 100000
#define N_EDGES 1600000
#define IN_C    128
#define OUT_C   64
#define SLOPE   0.2f

typedef float v2f __attribute__((ext_vector_type(2)));
typedef float v8f __attribute__((ext_vector_type(8)));

__device__ __forceinline__ float lrelu(float v) { return v > 0.f ? v : SLOPE * v; }

// Float atomic max via monotonic int/uint mapping (no NaNs expected).
// Lowers to GLOBAL_ATOMIC_MAX_I32 / GLOBAL_ATOMIC_MIN_U32 (native, no CAS loop).
__device__ __forceinline__ void atomicMaxF32(float* addr, float val) {
    if (val >= 0.f) atomicMax((int*)addr, __float_as_int(val));
    else            atomicMin((unsigned int*)addr, __float_as_uint(val));
}

// ---------------------------------------------------------------------------
// K1: feat[N,64] = x[N,128] @ W[128,64] + b   via V_WMMA_F32_16X16X4_F32
// One wave -> one 16x16 output tile. 32 WMMA issues (K=128 in steps of 4).
// A layout (ISA 7.12.2, 32-bit A 16x4): lanes 0-15: {K0,K1}, lanes 16-31: {K2,K3}
// B layout (4x16, row-striped): VGPR0: K0 (lanes0-15) / K2 (lanes16-31); VGPR1: K1/K3
// D layout: VGPR r: (M=r, N=lane) lanes0-15; (M=r+8, N=lane-16) lanes16-31
// ---------------------------------------------------------------------------
__global__ void __launch_bounds__(256)
k_gemm_feat(const float* __restrict__ x, const float* __restrict__ W,
            const float* __restrict__ bias, float* __restrict__ feat)
{
    const int lane = threadIdx.x & 31;
    const int wave = threadIdx.x >> 5;
    const int gw   = blockIdx.x * 8 + wave;        // tile id, 0..24999 (exact)
    const int mtile = gw >> 2;                     // waves 0..3 of a pair share mtile -> L1 reuse of x
    const int ntile = gw & 3;
    const int row0 = mtile * 16;
    const int col0 = ntile * 16;
    const int half = lane >> 4;                    // 0 -> K{0,1}, 1 -> K{2,3}
    const int l    = lane & 15;

    v8f acc = {};
    const float* xrow = x + (size_t)(row0 + l) * IN_C + 2 * half;       // contiguous float2 per step
    const float* wcol = W + (size_t)(2 * half) * OUT_C + col0 + l;

#pragma unroll 4
    for (int k = 0; k < IN_C; k += 4) {
        v2f a;
        a.x = xrow[k + 0];
        a.y = xrow[k + 1];
        v2f bb;
        bb.x = wcol[(size_t)k * OUT_C];
        bb.y = wcol[(size_t)(k + 1) * OUT_C];
        // (neg_a, A, neg_b, B, c_mod, C, reuse_a, reuse_b)
        acc = __builtin_amdgcn_wmma_f32_16x16x4_f32(false, a, false, bb,
                                                    (short)0, acc, false, false);
    }

    const float bv = bias[col0 + l];
    float* op = feat + (size_t)(row0 + 8 * half) * OUT_C + col0 + l;
#pragma unroll
    for (int r = 0; r < 8; ++r)
        op[(size_t)r * OUT_C] = acc[r] + bv;
}

// ---------------------------------------------------------------------------
// K2: per node: e0,e1 = feat[n] . att[:,0/1]; raw_self; m init; denom init
// ---------------------------------------------------------------------------
__global__ void __launch_bounds__(256)
k_node_pre(const float* __restrict__ feat, const float* __restrict__ att,
           float* __restrict__ e0, float* __restrict__ e1,
           float* __restrict__ raw_self, float* __restrict__ m,
           float* __restrict__ denom)
{
    int n = blockIdx.x * blockDim.x + threadIdx.x;
    if (n >= N_NODES) return;
    const float4* f4 = (const float4*)(feat + (size_t)n * OUT_C);
    float s0 = 0.f, s1 = 0.f;
#pragma unroll
    for (int q = 0; q < OUT_C / 4; ++q) {
        float4 f = f4[q];
        int c = q * 4;
        s0 += f.x * att[2*(c+0)] + f.y * att[2*(c+1)] + f.z * att[2*(c+2)] + f.w * att[2*(c+3)];
        s1 += f.x * att[2*(c+0)+1] + f.y * att[2*(c+1)+1] + f.z * att[2*(c+2)+1] + f.w * att[2*(c+3)+1];
    }
    e0[n] = s0;
    e1[n] = s1;
    float rs = lrelu(s0 + s1);
    raw_self[n] = rs;
    m[n] = rs;          // max(seg_max, raw_self): seed with self; empty segments handled
    denom[n] = 0.f;
}

// ---------------------------------------------------------------------------
// K3: per edge: raw = lrelu(e0[tar] + e1[src]); atomic max into m[tar]
// ---------------------------------------------------------------------------
__global__ void __launch_bounds__(256)
k_edge_logit(const int* __restrict__ tar, const int* __restrict__ src,
             const float* __restrict__ e0, const float* __restrict__ e1,
             float* __restrict__ edgebuf, float* __restrict__ m)
{
    int e = blockIdx.x * blockDim.x + threadIdx.x;
    if (e >= N_EDGES) return;
    int t = tar[e], s = src[e];
    float r = lrelu(e0[t] + e1[s]);
    edgebuf[e] = r;
    atomicMaxF32(&m[t], r);
}

// ---------------------------------------------------------------------------
// K4e: per edge: ex = exp(raw - m[tar]); denom[tar] += ex   (in-place buffer)
// ---------------------------------------------------------------------------
__global__ void __launch_bounds__(256)
k_edge_exp(const int* __restrict__ tar, float* __restrict__ edgebuf,
           const float* __restrict__ m, float* __restrict__ denom)
{
    int e = blockIdx.x * blockDim.x + threadIdx.x;
    if (e >= N_EDGES) return;
    int t = tar[e];
    float ex = expf(edgebuf[e] - m[t]);
    edgebuf[e] = ex;
    unsafeAtomicAdd(&denom[t], ex);   // native GLOBAL_ATOMIC_ADD_F32
}

// ---------------------------------------------------------------------------
// K4n: per node: es = exp(raw_self - m); denom += es (no contention post-K4e)
// ---------------------------------------------------------------------------
__global__ void __launch_bounds__(256)
k_node_exp(const float* __restrict__ raw_self, const float* __restrict__ m,
           float* __restrict__ es, float* __restrict__ denom)
{
    int n = blockIdx.x * blockDim.x + threadIdx.x;
    if (n >= N_NODES) return;
    float v = expf(raw_self[n] - m[n]);
    es[n] = v;
    denom[n] = denom[n] + v;
}

// ---------------------------------------------------------------------------
// K5: out = alpha_self * feat (initializes d_out); 4 channels / thread
// ---------------------------------------------------------------------------
__global__ void __launch_bounds__(256)
k_self_out(const float* __restrict__ feat, const float* __restrict__ es,
           const float* __restrict__ denom, float* __restrict__ out)
{
    int tid = blockIdx.x * blockDim.x + threadIdx.x;
    if (tid >= N_NODES * (OUT_C / 4)) return;
    int n = tid >> 4;
    int q = tid & 15;
    float a = es[n] / denom[n];
    float4 f = ((const float4*)feat)[(size_t)n * 16 + q];
    float4 o;
    o.x = a * f.x; o.y = a * f.y; o.z = a * f.z; o.w = a * f.w;
    ((float4*)out)[(size_t)n * 16 + q] = o;
}

// ---------------------------------------------------------------------------
// K6: edge aggregation: out[tar] += (ex/denom[tar]) * feat[src]
// 16 lanes per edge, float4 gather + 4 native f32 atomic adds per lane
// ---------------------------------------------------------------------------
__global__ void __launch_bounds__(256)
k_edge_agg(const int* __restrict__ tar, const int* __restrict__ src,
           const float* __restrict__ edgebuf, const float* __restrict__ denom,
           const float* __restrict__ feat, float* __restrict__ out)
{
    int tid = blockIdx.x * blockDim.x + threadIdx.x;   // E*16 = 25.6M fits in int
    int e = tid >> 4;
    if (e >= N_EDGES) return;
    int q = tid & 15;
    int t = tar[e], s = src[e];
    float alpha = edgebuf[e] / denom[t];
    float4 f = ((const float4*)feat)[(size_t)s * 16 + q];
    float* ob = out + (size_t)t * OUT_C + q * 4;
    unsafeAtomicAdd(ob + 0, alpha * f.x);
    unsafeAtomicAdd(ob + 1, alpha * f.y);
    unsafeAtomicAdd(ob + 2, alpha * f.z);
    unsafeAtomicAdd(ob + 3, alpha * f.w);
}

// ---------------------------------------------------------------------------
extern "C" void kernel_launch(void* const* d_in, const int* in_sizes, int n_in,
                              void* d_out, int out_size, void* d_ws, size_t ws_size,
                              hipStream_t stream) {
    const float* x    = (const float*)d_in[0];           // [N,128]
    const int*   ei   = (const int*)d_in[1];             // [2,E] row0=tar row1=src
    const float* W    = (const float*)d_in[2];           // [128,64]
    const float* bias = (const float*)d_in[3];           // [64]
    const float* att  = (const float*)d_in[4];           // [64,2]
    float* out = (float*)d_out;                          // [N,64]

    const int* tar = ei;
    const int* src = ei + N_EDGES;

    // workspace layout (floats): feat | e0 | e1 | raw_self | m | denom | es | edgebuf
    float* feat     = (float*)d_ws;
    float* e0       = feat + (size_t)N_NODES * OUT_C;
    float* e1       = e0 + N_NODES;
    float* raw_self = e1 + N_NODES;
    float* m        = raw_self + N_NODES;
    float* denom    = m + N_NODES;
    float* es       = denom + N_NODES;
    float* edgebuf  = es + N_NODES;

    const int B = 256;

    // K1: GEMM via WMMA fp32. 25000 tiles, 8 waves/block -> 3125 blocks exact.
    k_gemm_feat<<<3125, B, 0, stream>>>(x, W, bias, feat);

    // K2: node preprocessing
    k_node_pre<<<(N_NODES + B - 1) / B, B, 0, stream>>>(feat, att, e0, e1, raw_self, m, denom);

    // K3: edge logits + segment max
    k_edge_logit<<<(N_EDGES + B - 1) / B, B, 0, stream>>>(tar, src, e0, e1, edgebuf, m);

    // K4: exponentials + segment sum
    k_edge_exp<<<(N_EDGES + B - 1) / B, B, 0, stream>>>(tar, edgebuf, m, denom);
    k_node_exp<<<(N_NODES + B - 1) / B, B, 0, stream>>>(raw_self, m, es, denom);

    // K5: self term initializes out
    k_self_out<<<(N_NODES * (OUT_C / 4) + B - 1) / B, B, 0, stream>>>(feat, es, denom, out);

    // K6: gather-weighted-scatter aggregation
    k_edge_agg<<<(N_EDGES * 16 + B - 1) / B, B, 0, stream>>>(tar, src, edgebuf, denom, feat, out);
}